// HamMD_69226282877035
// MI455X (gfx1250) — compile-verified
//
#include <hip/hip_runtime.h>
#include <math.h>

#define WB 32     // batches
#define DD 512    // d
#define NN 4096   // n
#define RR 64     // codebook size r
#define LP 68     // LDS row pitch in floats (bank-conflict-free, 16B aligned)

typedef float v2f __attribute__((ext_vector_type(2)));
typedef float v8f __attribute__((ext_vector_type(8)));
typedef int   v4i __attribute__((ext_vector_type(4)));

// ---- CDNA5 async global->LDS staging (ASYNCcnt-tracked), with fallback ----
#if defined(__has_builtin)
#if __has_builtin(__builtin_amdgcn_global_load_async_to_lds_b128)
#define USE_ASYNC_LDS 1
#endif
#endif

#if defined(USE_ASYNC_LDS)
typedef __attribute__((address_space(1))) v4i* gptr_b128;   // global int4*
typedef __attribute__((address_space(3))) v4i* lptr_b128;   // LDS int4*
#endif

// D = A(16x4 f32) * B(4x16 f32) + C(16x16 f32)  -- fp32 WMMA, K=4
__device__ __forceinline__ v8f wmma4(v2f a, v2f b, v8f c) {
  return __builtin_amdgcn_wmma_f32_16x16x4_f32(
      /*neg_a=*/false, a, /*neg_b=*/false, b,
      /*c_mod=*/(short)0, c, /*reuse_a=*/false, /*reuse_b=*/false);
}

// Stage a 64-row x 64-float tile (row i at gbase + i*grow_stride) into LDS
// with pitch LP. 256 threads; async path: 4x global_load_async_to_lds_b128
// per thread (same INST_OFFSET applies to both global and LDS address).
__device__ __forceinline__ void stage_tile64(const float* __restrict__ gbase,
                                             int grow_stride, float* s, int t) {
  const int r = t >> 2;             // 0..63
  const int c = (t & 3) << 4;       // 0,16,32,48
  const float* gp = gbase + (size_t)r * grow_stride + c;
  float* sp = s + r * LP + c;
#if defined(USE_ASYNC_LDS)
  float* gpnc = const_cast<float*>(gp);
  gptr_b128 g = (gptr_b128)gpnc;
  lptr_b128 l = (lptr_b128)sp;
  __builtin_amdgcn_global_load_async_to_lds_b128(g, l, 0, 0);
  __builtin_amdgcn_global_load_async_to_lds_b128(g, l, 16, 0);
  __builtin_amdgcn_global_load_async_to_lds_b128(g, l, 32, 0);
  __builtin_amdgcn_global_load_async_to_lds_b128(g, l, 48, 0);
#else
  const float4* g = reinterpret_cast<const float4*>(gp);
  float4* d = reinterpret_cast<float4*>(sp);
  d[0] = g[0]; d[1] = g[1]; d[2] = g[2]; d[3] = g[3];
#endif
}

__device__ __forceinline__ void async_wait0() {
#if defined(USE_ASYNC_LDS)
  asm volatile("s_wait_asynccnt 0x0" ::: "memory");
#endif
}

// ---------------------------------------------------------------------------
// Kernel 1: cos = (D^T X) * invnorm(col), softmax over r, write C, accum sums
// grid (NN/64, WB), block 256 (8 waves). Double-buffered async staging.
// ---------------------------------------------------------------------------
__global__ __launch_bounds__(256) void vq_cos_softmax(
    const float* __restrict__ X, const float* __restrict__ D,
    float* __restrict__ C, float* __restrict__ sums) {
  __shared__ float sDb[2][64 * LP];
  __shared__ float sXb[2][64 * LP];
  __shared__ float colSSq[4 * 64];
  __shared__ float colInv[64];
  __shared__ float colMax[64];
  __shared__ float colInvSum[64];
  __shared__ float sRow[64];

  const int t = threadIdx.x;
  const int b = blockIdx.y;
  const int n0 = blockIdx.x * 64;
  const int wave = t >> 5, lane = t & 31, half = lane >> 4, l16 = lane & 15;

  if (t < 64) sRow[t] = 0.f;
  colSSq[t] = 0.f;

  const int id0 = wave * 2;  // each wave owns 2 of the 16 output tiles (4x4)
  v8f acc0 = {}; v8f acc1 = {};

  const float* Xb = X + (size_t)b * DD * NN + n0;
  const float* Db = D + (size_t)b * DD * RR;

  // prologue: stage chunk 0
  stage_tile64(Db, RR, sDb[0], t);                 // sD[k][r]
  stage_tile64(Xb, NN, sXb[0], t);                 // sX[k][col]

  for (int kc = 0; kc < DD / 64; ++kc) {
    async_wait0();
    __syncthreads();          // chunk kc resident; chunk kc-1 compute retired
    if (kc + 1 < DD / 64) {   // overlap: stream chunk kc+1 while computing kc
      stage_tile64(Db + (size_t)(kc + 1) * 64 * RR, RR, sDb[(kc + 1) & 1], t);
      stage_tile64(Xb + (size_t)(kc + 1) * 64 * NN, NN, sXb[(kc + 1) & 1], t);
    }
    const float* sD = sDb[kc & 1];
    const float* sX = sXb[kc & 1];
    // column sum-of-squares partials (spread over all 256 threads)
    {
      const int q = t >> 6, c = t & 63;
      float ss = 0.f;
      for (int i = 0; i < 16; ++i) { float v = sX[(q * 16 + i) * LP + c]; ss += v * v; }
      colSSq[q * 64 + c] += ss;
    }
    // WMMA: A = D^T (r x d), B = X (d x n)
    for (int ks = 0; ks < 16; ++ks) {
      const int k = ks * 4 + half * 2;
      {
        const int mi = id0 >> 2, ni = id0 & 3;
        v2f a, bb;
        a[0]  = sD[(k + 0) * LP + mi * 16 + l16];
        a[1]  = sD[(k + 1) * LP + mi * 16 + l16];
        bb[0] = sX[(k + 0) * LP + ni * 16 + l16];
        bb[1] = sX[(k + 1) * LP + ni * 16 + l16];
        acc0 = wmma4(a, bb, acc0);
      }
      {
        const int id = id0 + 1, mi = id >> 2, ni = id & 3;
        v2f a, bb;
        a[0]  = sD[(k + 0) * LP + mi * 16 + l16];
        a[1]  = sD[(k + 1) * LP + mi * 16 + l16];
        bb[0] = sX[(k + 0) * LP + ni * 16 + l16];
        bb[1] = sX[(k + 1) * LP + ni * 16 + l16];
        acc1 = wmma4(a, bb, acc1);
      }
    }
  }
  __syncthreads();
  if (t < 64) {
    const float ss = colSSq[t] + colSSq[64 + t] + colSSq[128 + t] + colSSq[192 + t];
    colInv[t] = 1.0f / fmaxf(sqrtf(ss), 1e-6f);   // Xn scale, applied post-GEMM
  }
  __syncthreads();
  // scatter scaled cos tiles into LDS (reuse retired sDb[0] buffer as sCos)
  float* sCos = &sDb[0][0];
  {
    int id = id0, mi = id >> 2, ni = id & 3;
    for (int v = 0; v < 8; ++v) {
      const int row = mi * 16 + half * 8 + v, col = ni * 16 + l16;
      sCos[row * LP + col] = acc0[v] * colInv[col];
    }
    id = id0 + 1; mi = id >> 2; ni = id & 3;
    for (int v = 0; v < 8; ++v) {
      const int row = mi * 16 + half * 8 + v, col = ni * 16 + l16;
      sCos[row * LP + col] = acc1[v] * colInv[col];
    }
  }
  __syncthreads();
  if (t < 64) {  // per-column softmax stats over r=64 (temperature 10)
    float m = -3.4e38f;
    for (int r = 0; r < 64; ++r) m = fmaxf(m, sCos[r * LP + t]);
    float s = 0.f;
    for (int r = 0; r < 64; ++r) s += __expf((sCos[r * LP + t] - m) * 0.1f);
    colMax[t] = m;
    colInvSum[t] = 1.0f / s;
  }
  __syncthreads();
  {
    const int col = t & 63, rg = t >> 6;
    float* Cg = C + (size_t)b * RR * NN + n0;
    for (int i = 0; i < 16; ++i) {
      const int r = rg * 16 + i;
      const float cv = __expf((sCos[r * LP + col] - colMax[col]) * 0.1f) * colInvSum[col];
      Cg[(size_t)r * NN + col] = cv;        // coalesced row write
      atomicAdd(&sRow[r], cv);              // ds_add_f32
    }
  }
  __syncthreads();
  if (t < 64) atomicAdd(&sums[b * RR + t], sRow[t]);
}

// ---------------------------------------------------------------------------
// Kernel 2: XCt[b] (512x64) = X[b] (512x4096) @ C[b]^T (4096x64)
// grid (DD/64, WB): each WG owns a 64x64 output tile, contracts all n in-WG.
// Double-buffered async staging.
// ---------------------------------------------------------------------------
__global__ __launch_bounds__(256) void vq_xct(
    const float* __restrict__ X, const float* __restrict__ C,
    float* __restrict__ XCt) {
  __shared__ float sXb[2][64 * LP];
  __shared__ float sCb[2][64 * LP];
  const int t = threadIdx.x;
  const int b = blockIdx.y;
  const int d0 = blockIdx.x * 64;
  const int wave = t >> 5, lane = t & 31, half = lane >> 4, l16 = lane & 15;
  const int id0 = wave * 2;
  v8f acc0 = {}, acc1 = {};

  const float* Xb = X + (size_t)(b * DD + d0) * NN;
  const float* Cb = C + (size_t)b * RR * NN;

  stage_tile64(Xb, NN, sXb[0], t);   // sX[d][k], k = n-chunk col
  stage_tile64(Cb, NN, sCb[0], t);   // sC[r][k]

  for (int nc = 0; nc < NN / 64; ++nc) {
    async_wait0();
    __syncthreads();
    if (nc + 1 < NN / 64) {
      stage_tile64(Xb + (nc + 1) * 64, NN, sXb[(nc + 1) & 1], t);
      stage_tile64(Cb + (nc + 1) * 64, NN, sCb[(nc + 1) & 1], t);
    }
    const float* sX = sXb[nc & 1];
    const float* sC = sCb[nc & 1];
    for (int ks = 0; ks < 16; ++ks) {
      const int k = ks * 4 + half * 2;
      {
        const int mi = id0 >> 2, ni = id0 & 3;
        v2f a, bb;
        a[0]  = sX[(mi * 16 + l16) * LP + k + 0];
        a[1]  = sX[(mi * 16 + l16) * LP + k + 1];
        bb[0] = sC[(ni * 16 + l16) * LP + k + 0];
        bb[1] = sC[(ni * 16 + l16) * LP + k + 1];
        acc0 = wmma4(a, bb, acc0);
      }
      {
        const int id = id0 + 1, mi = id >> 2, ni = id & 3;
        v2f a, bb;
        a[0]  = sX[(mi * 16 + l16) * LP + k + 0];
        a[1]  = sX[(mi * 16 + l16) * LP + k + 1];
        bb[0] = sC[(ni * 16 + l16) * LP + k + 0];
        bb[1] = sC[(ni * 16 + l16) * LP + k + 1];
        acc1 = wmma4(a, bb, acc1);
      }
    }
  }
  {
    int id = id0, mi = id >> 2, ni = id & 3;
    float* O = XCt + (size_t)(b * DD + d0) * RR;
    for (int v = 0; v < 8; ++v)
      O[(size_t)(mi * 16 + half * 8 + v) * RR + ni * 16 + l16] = acc0[v];
    id = id0 + 1; mi = id >> 2; ni = id & 3;
    for (int v = 0; v < 8; ++v)
      O[(size_t)(mi * 16 + half * 8 + v) * RR + ni * 16 + l16] = acc1[v];
  }
}

// ---------------------------------------------------------------------------
// Kernel 3: Dnew = normalize((XCt / max(sums,eps)), axis=d). 2048 columns.
// ---------------------------------------------------------------------------
__global__ __launch_bounds__(256) void vq_dnew(
    const float* __restrict__ XCt, const float* __restrict__ sums,
    float* __restrict__ D) {
  const int g = blockIdx.x * 256 + threadIdx.x;   // 0..2047 = b*64 + r
  const int b = g >> 6, r = g & 63;
  const float s = fmaxf(sums[g], 1e-6f);
  const float invs = 1.0f / s;
  const float* col = XCt + (size_t)b * DD * RR + r;
  float ss = 0.f;
  for (int dd = 0; dd < DD; ++dd) { const float v = col[dd * RR] * invs; ss += v * v; }
  const float scl = invs / fmaxf(sqrtf(ss), 1e-6f);
  float* Dc = D + (size_t)b * DD * RR + r;
  for (int dd = 0; dd < DD; ++dd) Dc[dd * RR] = col[dd * RR] * scl;
}

// Initial D = normalize(D_init, axis=d)
__global__ __launch_bounds__(256) void vq_init_d(
    const float* __restrict__ Din, float* __restrict__ D) {
  const int g = blockIdx.x * 256 + threadIdx.x;
  const int b = g >> 6, r = g & 63;
  const float* col = Din + (size_t)b * DD * RR + r;
  float ss = 0.f;
  for (int dd = 0; dd < DD; ++dd) { const float v = col[dd * RR]; ss += v * v; }
  const float scl = 1.0f / fmaxf(sqrtf(ss), 1e-6f);
  float* Dc = D + (size_t)b * DD * RR + r;
  for (int dd = 0; dd < DD; ++dd) Dc[dd * RR] = col[dd * RR] * scl;
}

// ---------------------------------------------------------------------------
// Kernel 4: Xbar[b] (512x4096) = Dnew[b] (512x64) @ C[b] (64x4096)
// grid (NN/64, WB). K = r = 64. A-frags hoisted across the 4 N-tiles.
// ---------------------------------------------------------------------------
__global__ __launch_bounds__(256) void vq_xbar(
    const float* __restrict__ D, const float* __restrict__ C,
    float* __restrict__ out) {
  __shared__ float sC[64 * LP];
  const int t = threadIdx.x;
  const int b = blockIdx.y;
  const int n0 = blockIdx.x * 64;
  const int wave = t >> 5, lane = t & 31, half = lane >> 4, l16 = lane & 15;

  const float* Db = D + (size_t)b * DD * RR;
  __builtin_prefetch(Db + (size_t)t * 2 * RR, 0, 1);   // warm D[b] (global_prefetch_b8)

  stage_tile64(C + (size_t)b * RR * NN + n0, NN, sC, t);   // sC[r][col]
  async_wait0();
  __syncthreads();

  float* Ob = out + (size_t)b * DD * NN + n0;

  for (int p = 0; p < 4; ++p) {
    const int mi = wave * 4 + p;   // d-tile 0..31
    v2f afr[16];
    const float2* Arow = reinterpret_cast<const float2*>(Db + (size_t)(mi * 16 + l16) * RR);
    for (int ks = 0; ks < 16; ++ks) {
      const float2 av = Arow[ks * 2 + half];   // A[m][k..k+1], k = ks*4 + half*2
      afr[ks][0] = av.x; afr[ks][1] = av.y;
    }
    for (int ni = 0; ni < 4; ++ni) {
      v8f acc = {};
      for (int ks = 0; ks < 16; ++ks) {
        const int k = ks * 4 + half * 2;
        v2f bb;
        bb[0] = sC[(k + 0) * LP + ni * 16 + l16];
        bb[1] = sC[(k + 1) * LP + ni * 16 + l16];
        acc = wmma4(afr[ks], bb, acc);
      }
      for (int v = 0; v < 8; ++v)
        Ob[(size_t)(mi * 16 + half * 8 + v) * NN + ni * 16 + l16] = acc[v];
    }
  }
}

// ---------------------------------------------------------------------------
extern "C" void kernel_launch(void* const* d_in, const int* in_sizes, int n_in,
                              void* d_out, int out_size, void* d_ws, size_t ws_size,
                              hipStream_t stream) {
  const float* X = (const float*)d_in[0];       // (32, 512, 4096)
  const float* Dinit = (const float*)d_in[1];   // (32, 512, 64)
  float* out = (float*)d_out;                   // (32, 512, 4096)

  float* ws = (float*)d_ws;
  float* wD    = ws;                                      // 32*512*64   = 4MB
  float* wC    = wD   + (size_t)WB * DD * RR;             // 32*64*4096  = 32MB
  float* wXCt  = wC   + (size_t)WB * RR * NN;             // 32*512*64   = 4MB
  float* wSums = wXCt + (size_t)WB * DD * RR;             // 32*64       = 8KB
  (void)ws_size; (void)in_sizes; (void)n_in; (void)out_size;

  vq_init_d<<<WB * RR / 256, 256, 0, stream>>>(Dinit, wD);

  for (int step = 0; step < 3; ++step) {
    (void)hipMemsetAsync(wSums, 0, (size_t)WB * RR * sizeof(float), stream);
    vq_cos_softmax<<<dim3(NN / 64, WB), 256, 0, stream>>>(X, wD, wC, wSums);
    vq_xct<<<dim3(DD / 64, WB), 256, 0, stream>>>(X, wC, wXCt);
    vq_dnew<<<WB * RR / 256, 256, 0, stream>>>(wXCt, wSums, wD);
  }
  // Final reconstruction with Dnew from the last step and its C
  vq_xbar<<<dim3(NN / 64, WB), 256, 0, stream>>>(wD, wC, out);
}